// RNADE_85469849190695
// MI455X (gfx1250) — compile-verified
//
#include <hip/hip_runtime.h>
#include <math.h>

#define BB 1024
#define DD 2048
#define HH 64
#define CHUNKS 64          // chunks along D
#define CLEN 32            // D per chunk (CHUNKS*CLEN == DD)

typedef __attribute__((ext_vector_type(2))) float v2f;
typedef __attribute__((ext_vector_type(8))) float v8f;

// ---------------------------------------------------------------------------
// Pass 1: chunk sums  S[b][ch][h] = sum_{k in chunk} x[b,k] * W[h,k]
// One wave per 16(b) x 16(h) output tile per chunk; K loop in steps of 4
// using V_WMMA_F32_16X16X4_F32 (f32 in / f32 acc -> matches reference math).
// S layout: S[(b*CHUNKS + ch)*HH + h]
// ---------------------------------------------------------------------------
__global__ void rnade_chunk_gemm(const float* __restrict__ X,
                                 const float* __restrict__ W,   // [H, D]
                                 float* __restrict__ S) {
    const unsigned wid  = (blockIdx.x * blockDim.x + threadIdx.x) >> 5;
    const unsigned lane = threadIdx.x & 31u;

    const unsigned chunk = wid & (CHUNKS - 1);       // 0..63
    const unsigned ht    = (wid >> 6) & 3;           // 0..3  (H/16)
    const unsigned bt    = wid >> 8;                 // 0..63 (B/16)

    const unsigned b0 = bt * 16, h0 = ht * 16, d0 = chunk * CLEN;
    const unsigned half = lane >> 4;                 // 0: K pair {0,1}; 1: {2,3}
    const unsigned lm   = lane & 15u;

    v8f acc = {0.f, 0.f, 0.f, 0.f, 0.f, 0.f, 0.f, 0.f};

    #pragma unroll
    for (unsigned kk = 0; kk < CLEN; kk += 4) {
        const unsigned k = d0 + kk + 2 * half;       // 8B aligned
        // A frag: 16x4 f32, lane<16 -> M=lane, K={k,k+1}; lane>=16 -> K={k+2,k+3}
        v2f a = *(const v2f*)(X + (b0 + lm) * DD + k);
        // B frag: 4x16 f32, lane<16 -> N=lane, K={k,k+1}
        v2f bf = *(const v2f*)(W + (h0 + lm) * DD + k);
        acc = __builtin_amdgcn_wmma_f32_16x16x4_f32(
                  false, a, false, bf, (short)0, acc, false, false);
    }

    // D layout: VGPR r, lanes 0-15 -> M=r, N=lane; lanes 16-31 -> M=r+8
    #pragma unroll
    for (int r = 0; r < 8; ++r) {
        const unsigned M = r + 8 * half;
        S[((b0 + M) * CHUNKS + chunk) * HH + h0 + lm] = acc[r];
    }
}

// ---------------------------------------------------------------------------
// Pass 2: in-place exclusive scan over chunks (+ bias c):
//   A0[b][ch][h] = c[h] + sum_{j<ch} S[b][j][h]     (overwrites S)
// ---------------------------------------------------------------------------
__global__ void rnade_chunk_scan(float* __restrict__ S,
                                 const float* __restrict__ c) {
    const unsigned t = blockIdx.x * blockDim.x + threadIdx.x;  // B*H threads
    const unsigned b = t >> 6, h = t & (HH - 1);
    float acc = c[h];
    float* p = S + b * CHUNKS * HH + h;
    #pragma unroll 4
    for (int ch = 0; ch < CHUNKS; ++ch) {
        float tmp = p[ch * HH];
        p[ch * HH] = acc;
        acc += tmp;
    }
}

// ---------------------------------------------------------------------------
// Pass 3: per-(b,chunk) sequential scan of length CLEN.
// Block = 256 threads = 8 waves: 8 consecutive b's, same chunk.
// Lane l owns state A[h=l], A[h=l+32].
// Chunk weights (W,P,V,V2 columns) staged in LDS; per-step dots reduced by
// wave32 butterfly; outputs buffered in LDS and stored coalesced.
// ---------------------------------------------------------------------------
__global__ void rnade_scan(const float* __restrict__ X,
                           const float* __restrict__ W,    // [H,D]
                           const float* __restrict__ P,    // [D,H]
                           const float* __restrict__ V,    // [D,H]
                           const float* __restrict__ V2,   // [D,H]
                           const float* __restrict__ bb,   // [D]
                           const float* __restrict__ b2,   // [D]
                           const float* __restrict__ A0,   // scanned S
                           float* __restrict__ out) {
    __shared__ float wsh[CLEN * 4 * HH];   // [d][arr(W,P,V,V2)][h]  32 KB
    __shared__ float xsh[8 * CLEN];        // x tile                  1 KB
    __shared__ float outb[8 * 2 * CLEN];   // per-wave dot results    2 KB

    const unsigned chunk = blockIdx.x & (CHUNKS - 1);
    const unsigned bgrp  = blockIdx.x >> 6;            // 0..127
    const unsigned wave  = threadIdx.x >> 5;           // 0..7
    const unsigned lane  = threadIdx.x & 31u;
    const unsigned b     = bgrp * 8 + wave;
    const unsigned d0    = chunk * CLEN;

    // ---- stage weights: thread (g,d): g = tid>>5 picks (arr, h-half) ----
    {
        const unsigned d   = threadIdx.x & (CLEN - 1);
        const unsigned g   = threadIdx.x >> 5;
        const unsigned arr = g >> 1;                   // 0..3, uniform/thread
        const float* Gp = (arr == 0) ? W : (arr == 1) ? P : (arr == 2) ? V : V2;
        const unsigned hb = (g & 1) * 32;
        #pragma unroll 8
        for (unsigned i = 0; i < 32; ++i) {
            const unsigned h = hb + i;
            // W is [H,D] row-major; P/V/V2 are [D,H]: index accordingly
            float val = (arr == 0) ? Gp[h * DD + d0 + d]
                                   : Gp[(d0 + d) * HH + h];
            wsh[d * 256 + arr * 64 + h] = val;
        }
        xsh[threadIdx.x & 255u] = X[(bgrp * 8 + (threadIdx.x >> 5)) * DD + d0 + d];
    }
    __syncthreads();

    // ---- load state ----
    const unsigned abase = (b * CHUNKS + chunk) * HH;
    float a0 = A0[abase + lane];
    float a1 = A0[abase + lane + 32];

    const unsigned obase = wave * 2 * CLEN;

    for (int dd = 0; dd < CLEN; ++dd) {
        const float xv = xsh[wave * CLEN + dd];
        const int wb = dd * 256;
        const float w0 = wsh[wb +   0 + lane], w1 = wsh[wb +  32 + lane];
        const float p0 = wsh[wb +  64 + lane], p1 = wsh[wb +  96 + lane];
        const float v0 = wsh[wb + 128 + lane], v1 = wsh[wb + 160 + lane];
        const float u0 = wsh[wb + 192 + lane], u1 = wsh[wb + 224 + lane];

        const float t0 = a0 * p0, t1 = a1 * p1;
        const float h0 = fmaxf(t0, 0.2f * t0);        // leaky_relu, slope<1
        const float h1 = fmaxf(t1, 0.2f * t1);

        float sV = h0 * v0 + h1 * v1;
        float sU = h0 * u0 + h1 * u1;
        #pragma unroll
        for (int m = 16; m >= 1; m >>= 1) {
            sV += __shfl_xor(sV, m, 32);
            sU += __shfl_xor(sU, m, 32);
        }
        if (lane == 0) {
            outb[obase + dd]        = sV;
            outb[obase + CLEN + dd] = sU;
        }
        a0 = fmaf(xv, w0, a0);                        // post-update (exclusive)
        a1 = fmaf(xv, w1, a1);
    }

    // ---- coalesced epilogue: lane l handles d = d0 + l ----
    const unsigned dg = d0 + lane;
    const float dV = outb[obase + lane];
    const float dU = outb[obase + CLEN + lane];
    const float alpha = 2.0f / (1.0f + __expf(-(dV + bb[dg]))) + 0.6f;
    const float mean  = dU + b2[dg];
    out[b * DD + dg]            = mean;               // [0] = mean
    out[BB * DD + b * DD + dg]  = alpha;              // [1] = alpha
}

// ---------------------------------------------------------------------------
extern "C" void kernel_launch(void* const* d_in, const int* in_sizes, int n_in,
                              void* d_out, int out_size, void* d_ws, size_t ws_size,
                              hipStream_t stream) {
    const float* x  = (const float*)d_in[0];
    const float* V  = (const float*)d_in[1];
    const float* bv = (const float*)d_in[2];
    const float* V2 = (const float*)d_in[3];
    const float* b2 = (const float*)d_in[4];
    const float* W  = (const float*)d_in[5];
    const float* c  = (const float*)d_in[6];
    const float* P  = (const float*)d_in[7];
    float* out = (float*)d_out;
    float* S   = (float*)d_ws;   // B*CHUNKS*H f32 = 16 MB

    // Pass 1: waves = (B/16)*(H/16)*CHUNKS = 64*4*64 = 16384 -> 2048 blocks
    rnade_chunk_gemm<<<2048, 256, 0, stream>>>(x, W, S);
    // Pass 2: B*H = 65536 threads
    rnade_chunk_scan<<<256, 256, 0, stream>>>(S, c);
    // Pass 3: blocks = (B/8)*CHUNKS = 128*64 = 8192
    rnade_scan<<<8192, 256, 0, stream>>>(x, W, P, V, V2, bv, b2, S, out);
}